// dotAttn_7705171329749
// MI455X (gfx1250) — compile-verified
//
#include <hip/hip_runtime.h>
#include <hip/hip_bf16.h>

typedef __attribute__((ext_vector_type(2))) float v2f;
typedef __attribute__((ext_vector_type(8))) float v8f;

#define S_LEN 512
#define D_LEN 1024
#define E_LEN 16
#define KB 16          // K-chunk staged in LDS per iteration
#define LDS_STRIDE 20  // 16 + 4 pad: 16B-aligned rows (80B), conflict-free

// ---------------------------------------------------------------------------
// Kernel 1: masked gather + sum-pool.  e1[b,d] = sum_e mask*embeds[b,idx,d]
// One block per batch, 256 threads x float4.
// ---------------------------------------------------------------------------
__global__ __launch_bounds__(256) void pool_kernel(
    const float* __restrict__ embeds,
    const int* __restrict__ e1_idx, const int* __restrict__ e2_idx,
    const int* __restrict__ e1_mask, const int* __restrict__ e2_mask,
    float* __restrict__ e1, float* __restrict__ e2)
{
    const int b = blockIdx.x;
    const int t = threadIdx.x;
    const int d = t * 4;
    const float* E = embeds + (size_t)b * S_LEN * D_LEN;

    float4 a1 = make_float4(0.f, 0.f, 0.f, 0.f);
    float4 a2 = make_float4(0.f, 0.f, 0.f, 0.f);
    #pragma unroll
    for (int e = 0; e < E_LEN; ++e) {
        const int i1 = e1_idx[b * E_LEN + e];
        const int i2 = e2_idx[b * E_LEN + e];
        const float m1 = (float)e1_mask[b * E_LEN + e];
        const float m2 = (float)e2_mask[b * E_LEN + e];
        const float4 g1 = *(const float4*)(E + (size_t)i1 * D_LEN + d);
        const float4 g2 = *(const float4*)(E + (size_t)i2 * D_LEN + d);
        a1.x = fmaf(m1, g1.x, a1.x); a1.y = fmaf(m1, g1.y, a1.y);
        a1.z = fmaf(m1, g1.z, a1.z); a1.w = fmaf(m1, g1.w, a1.w);
        a2.x = fmaf(m2, g2.x, a2.x); a2.y = fmaf(m2, g2.y, a2.y);
        a2.z = fmaf(m2, g2.z, a2.z); a2.w = fmaf(m2, g2.w, a2.w);
    }
    *(float4*)(e1 + (size_t)b * D_LEN + d) = a1;
    *(float4*)(e2 + (size_t)b * D_LEN + d) = a2;
}

// ---------------------------------------------------------------------------
// Kernel 2: WMMA batched dual mat-vec + softmax.
// One block (256 threads = 8 waves) per batch. Wave w owns rows [64w, 64w+64)
// as 4 tiles of 16.  A = 16x4 f32 embeds fragment (from LDS slab),
// B = 4x16 with col0=e1 chunk, col1=e2 chunk, rest 0.
// Slab staging uses the CDNA5 async global->LDS path (no VGPR round-trip).
// ---------------------------------------------------------------------------
__global__ __launch_bounds__(256) void attn_kernel(
    const float* __restrict__ embeds,
    const float* __restrict__ e1v, const float* __restrict__ e2v,
    float* __restrict__ out)
{
    __shared__ float tile[S_LEN * LDS_STRIDE];  // 512*20*4 = 40 KB
    __shared__ float sc1[S_LEN];
    __shared__ float sc2[S_LEN];
    __shared__ float red[256];

    const int b     = blockIdx.x;
    const int t     = threadIdx.x;
    const int lane  = t & 31;
    const int wave  = t >> 5;
    const int laneN = lane & 15;   // N-index / M-row within half-wave
    const int half  = lane >> 4;   // 0: K=0,1   1: K=2,3

    const float* E  = embeds + (size_t)b * S_LEN * D_LEN;
    const float* v1 = e1v + (size_t)b * D_LEN;
    const float* v2 = e2v + (size_t)b * D_LEN;

    v8f acc[4];
    #pragma unroll
    for (int j = 0; j < 4; ++j) acc[j] = (v8f){0.f,0.f,0.f,0.f,0.f,0.f,0.f,0.f};

    const int colq = (t & 3) * 4;  // 0,4,8,12
    const int row0 = t >> 2;       // 0..63

    // LDS aperture maps the byte offset into addr[31:0]; truncation of the
    // generic pointer yields the wave-relative LDS byte address.
    const unsigned lds_base = (unsigned)(size_t)(&tile[0]);

    for (int kb = 0; kb < D_LEN; kb += KB) {
        __syncthreads();  // all waves done reading the previous slab
        // ---- async-stage [512 x KB] slab: global -> LDS, no VGPR transit ----
        #pragma unroll
        for (int p = 0; p < 8; ++p) {
            const int row = row0 + p * 64;
            const unsigned ldsoff =
                lds_base + (unsigned)(row * LDS_STRIDE + colq) * 4u;
            const float* gp = E + (size_t)row * D_LEN + kb + colq;
            asm volatile("global_load_async_to_lds_b128 %0, %1, off"
                         :: "v"(ldsoff), "v"(gp) : "memory");
        }
        asm volatile("s_wait_asynccnt 0x0" ::: "memory");  // own writes landed
        __syncthreads();                                   // slab published
        if (kb + KB < D_LEN) {  // uniform branch: speculative prefetch of next slab
            __builtin_prefetch(E + (size_t)row0 * D_LEN + kb + KB + colq, 0, 1);
        }
        // ---- 4 K-steps of 4 per chunk; 4 row-tiles per wave ----
        #pragma unroll
        for (int k = 0; k < KB; k += 4) {
            const int dk = kb + k + 2 * half;
            // B fragment: col0 = e1, col1 = e2, others 0 (cndmask selects)
            const float b0 = (laneN == 0) ? v1[dk]     : (laneN == 1) ? v2[dk]     : 0.f;
            const float b1 = (laneN == 0) ? v1[dk + 1] : (laneN == 1) ? v2[dk + 1] : 0.f;
            const v2f bf = {b0, b1};
            #pragma unroll
            for (int j = 0; j < 4; ++j) {
                const int r = wave * 64 + j * 16 + laneN;
                const float* ap = &tile[r * LDS_STRIDE + k + 2 * half];
                const v2f af = {ap[0], ap[1]};
                acc[j] = __builtin_amdgcn_wmma_f32_16x16x4_f32(
                    /*neg_a=*/false, af, /*neg_b=*/false, bf,
                    /*c_mod=*/(short)0, acc[j], /*reuse_a=*/false, /*reuse_b=*/false);
            }
        }
    }

    // ---- extract score columns 0 (e1) and 1 (e2) from C layout ----
    // lane n<16: VGPR r = C[M=r][N=n]; lane 16+n: VGPR r = C[M=8+r][N=n]
    if (laneN < 2) {
        float* dstarr = (laneN == 0) ? sc1 : sc2;
        #pragma unroll
        for (int j = 0; j < 4; ++j) {
            const int base = wave * 64 + j * 16 + half * 8;
            #pragma unroll
            for (int r = 0; r < 8; ++r) dstarr[base + r] = acc[j][r];
        }
    }
    __syncthreads();

    // ---- softmax over S=512 (each thread owns elements t and t+256) ----
    const float x1a = sc1[t], x1b = sc1[t + 256];
    const float x2a = sc2[t], x2b = sc2[t + 256];

    red[t] = fmaxf(x1a, x1b); __syncthreads();
    for (int off = 128; off > 0; off >>= 1) {
        if (t < off) red[t] = fmaxf(red[t], red[t + off]);
        __syncthreads();
    }
    const float gmax1 = red[0]; __syncthreads();

    red[t] = fmaxf(x2a, x2b); __syncthreads();
    for (int off = 128; off > 0; off >>= 1) {
        if (t < off) red[t] = fmaxf(red[t], red[t + off]);
        __syncthreads();
    }
    const float gmax2 = red[0]; __syncthreads();

    const float p1a = __expf(x1a - gmax1), p1b = __expf(x1b - gmax1);
    const float p2a = __expf(x2a - gmax2), p2b = __expf(x2b - gmax2);

    red[t] = p1a + p1b; __syncthreads();
    for (int off = 128; off > 0; off >>= 1) {
        if (t < off) red[t] += red[t + off];
        __syncthreads();
    }
    const float sum1 = red[0]; __syncthreads();

    red[t] = p2a + p2b; __syncthreads();
    for (int off = 128; off > 0; off >>= 1) {
        if (t < off) red[t] += red[t + off];
        __syncthreads();
    }
    const float sum2 = red[0];

    float* O = out + (size_t)b * S_LEN;
    O[t]       = 0.5f * (p1a / sum1 + p2a / sum2);
    O[t + 256] = 0.5f * (p1b / sum1 + p2b / sum2);
}

// ---------------------------------------------------------------------------
extern "C" void kernel_launch(void* const* d_in, const int* in_sizes, int n_in,
                              void* d_out, int out_size, void* d_ws, size_t ws_size,
                              hipStream_t stream) {
    const float* embeds = (const float*)d_in[0];
    const int* e1_idx   = (const int*)d_in[1];
    const int* e2_idx   = (const int*)d_in[2];
    const int* e1_mask  = (const int*)d_in[3];
    const int* e2_mask  = (const int*)d_in[4];
    float* out = (float*)d_out;

    const int B = 64;
    float* e1 = (float*)d_ws;            // [B, D]
    float* e2 = e1 + (size_t)B * D_LEN;  // [B, D]

    pool_kernel<<<B, 256, 0, stream>>>(embeds, e1_idx, e2_idx, e1_mask, e2_mask, e1, e2);
    attn_kernel<<<B, 256, 0, stream>>>(embeds, e1, e2, out);
}